// VariationalGCNEncoder_10282151706739
// MI455X (gfx1250) — compile-verified
//
#include <hip/hip_runtime.h>

typedef __attribute__((ext_vector_type(2))) float v2f;
typedef __attribute__((ext_vector_type(8))) float v8f;

#define D_OUT 128
#define MAXK 256
#define PADK 4

// ---------------------------------------------------------------------------
// Degree / normalization: deg[i] = 1 (self loop) + #edges with dst==i,
// then dinv[i] = rsqrt(deg[i]) (deg >= 1 always, matches reference guard).
// ---------------------------------------------------------------------------
__global__ __launch_bounds__(256) void init_deg_kernel(float* __restrict__ deg, int n) {
    int i = blockIdx.x * 256 + threadIdx.x;
    if (i < n) deg[i] = 1.0f;
}

__global__ __launch_bounds__(256) void accum_deg_kernel(float* __restrict__ deg,
                                                        const int* __restrict__ dst, int e) {
    int i = blockIdx.x * 256 + threadIdx.x;
    if (i < e) atomicAdd(&deg[dst[i]], 1.0f);
}

__global__ __launch_bounds__(256) void dinv_kernel(float* __restrict__ deg, int n) {
    int i = blockIdx.x * 256 + threadIdx.x;
    if (i < n) deg[i] = __frsqrt_rn(deg[i]);
}

// ---------------------------------------------------------------------------
// C[N x 128] = A[N x K] * B[K x 128], fp32 WMMA (V_WMMA_F32_16X16X4_F32).
// One block = 8 waves = one 16-row strip; wave w owns column tile w.
// A strip staged in LDS with padded stride (conflict-free fragment reads).
// K is a power of two (256 or 128); kshift = log2(K).
// ---------------------------------------------------------------------------
__global__ __launch_bounds__(256)
void gemm_wmma_f32_kernel(const float* __restrict__ A, const float* __restrict__ B,
                          float* __restrict__ C, int N, int K, int kshift) {
    __shared__ float lds_a[16 * (MAXK + PADK)];
    const int ldk  = K + PADK;
    const int row0 = blockIdx.x * 16;
    const int tid  = threadIdx.x;
    const int kmask = K - 1;

    // Cooperative, coalesced load of the 16 x K strip of A into LDS.
    for (int idx = tid; idx < 16 * K; idx += 256) {
        int r = idx >> kshift;
        int c = idx & kmask;
        int row = row0 + r;
        lds_a[r * ldk + c] = (row < N) ? A[(size_t)row * K + c] : 0.0f;
    }
    __syncthreads();

    const int wave = tid >> 5;
    const int lane = tid & 31;
    const int n0   = wave * 16;        // this wave's output column tile
    const int mlo  = lane & 15;
    const int hi   = lane >> 4;        // 0: K offsets {0,1}, 1: K offsets {2,3}
    const int ksel = hi << 1;

    const float* arow = &lds_a[mlo * ldk];   // A fragment row (M = lane&15)
    v8f acc = {};
    for (int k0 = 0; k0 < K; k0 += 4) {
        v2f a, b;
        a[0] = arow[k0 + ksel];
        a[1] = arow[k0 + ksel + 1];
        b[0] = B[(size_t)(k0 + ksel)     * D_OUT + n0 + mlo];
        b[1] = B[(size_t)(k0 + ksel + 1) * D_OUT + n0 + mlo];
        acc = __builtin_amdgcn_wmma_f32_16x16x4_f32(
            /*neg_a=*/false, a, /*neg_b=*/false, b,
            /*c_mod=*/(short)0, acc, /*reuse_a=*/false, /*reuse_b=*/false);
    }

    // C/D layout: VGPR r -> row (r + 8*hi), col = n0 + (lane&15)
    const int col   = n0 + mlo;
    const int rbase = row0 + hi * 8;
#pragma unroll
    for (int r = 0; r < 8; ++r) {
        int row = rbase + r;
        if (row < N) C[(size_t)row * D_OUT + col] = acc[r];
    }
}

// ---------------------------------------------------------------------------
// out[i,:] = p[i,:] * dinv[i]^2 + bias[:]   (self-loop term + bias, pre-scatter)
// ---------------------------------------------------------------------------
__global__ __launch_bounds__(256)
void init_scaled_bias_kernel(float* __restrict__ out, const float* __restrict__ p,
                             const float* __restrict__ dinv, const float* __restrict__ bias,
                             int total) {
    int g = blockIdx.x * 256 + threadIdx.x;
    if (g >= total) return;
    int i = g >> 7;          // node
    int d = g & (D_OUT - 1); // feature
    float di = dinv[i];
    out[g] = p[g] * di * di + bias[d];
}

// ---------------------------------------------------------------------------
// Edge scatter: one wave per edge; lane l moves float4 of columns [4l..4l+3].
// out[dst,:] += p[src,:] * (dinv[src]*dinv[dst])  via global_atomic_add_f32.
// Gather read is a fully coalesced 512B row per wave.
// ---------------------------------------------------------------------------
__global__ __launch_bounds__(256)
void edge_scatter_kernel(float* __restrict__ out, const float* __restrict__ p,
                         const int* __restrict__ src, const int* __restrict__ dst,
                         const float* __restrict__ dinv, int e) {
    int g = blockIdx.x * 256 + threadIdx.x;
    int edge = g >> 5;
    if (edge >= e) return;
    int lane = g & 31;
    int s  = src[edge];
    int dt = dst[edge];
    float nrm = dinv[s] * dinv[dt];
    const float4 v = ((const float4*)(p + (size_t)s * D_OUT))[lane];
    float* o = out + (size_t)dt * D_OUT + lane * 4;
    atomicAdd(o + 0, v.x * nrm);
    atomicAdd(o + 1, v.y * nrm);
    atomicAdd(o + 2, v.z * nrm);
    atomicAdd(o + 3, v.w * nrm);
}

__global__ __launch_bounds__(256) void relu_kernel(float* __restrict__ h, int total) {
    int g = blockIdx.x * 256 + threadIdx.x;
    if (g < total) h[g] = fmaxf(h[g], 0.0f);
}

// ---------------------------------------------------------------------------
// Launcher
// ---------------------------------------------------------------------------
extern "C" void kernel_launch(void* const* d_in, const int* in_sizes, int n_in,
                              void* d_out, int out_size, void* d_ws, size_t ws_size,
                              hipStream_t stream) {
    const float* x   = (const float*)d_in[0];
    const int*   ei  = (const int*)  d_in[1];
    const float* W1  = (const float*)d_in[2];
    const float* b1  = (const float*)d_in[3];
    const float* Wmu = (const float*)d_in[4];
    const float* bmu = (const float*)d_in[5];
    const float* Wls = (const float*)d_in[6];
    const float* bls = (const float*)d_in[7];

    const int D_IN = 256;
    const int N = in_sizes[0] / D_IN;   // 100000
    const int E = in_sizes[1] / 2;      // 1600000
    const int* src = ei;
    const int* dst = ei + E;
    const int total = N * D_OUT;

    // Workspace: dinv | bufA (t1 / p_mu / p_ls) | bufB (h)
    char* ws = (char*)d_ws;
    size_t off = 0;
    float* dinv = (float*)(ws + off); off += (((size_t)N * 4) + 255) & ~(size_t)255;
    float* bufA = (float*)(ws + off); off += (((size_t)total * 4) + 255) & ~(size_t)255;
    float* bufB = (float*)(ws + off);

    float* mu = (float*)d_out;
    float* ls = mu + (size_t)total;

    dim3 blk(256);
    dim3 gN((N + 255) / 256);
    dim3 gE((E + 255) / 256);
    dim3 gT((total + 255) / 256);
    dim3 gScat((E * 32 + 255) / 256);
    dim3 gGemm((N + 15) / 16);

    // --- normalization ---
    init_deg_kernel<<<gN, blk, 0, stream>>>(dinv, N);
    accum_deg_kernel<<<gE, blk, 0, stream>>>(dinv, dst, E);
    dinv_kernel<<<gN, blk, 0, stream>>>(dinv, N);

    // --- layer 1: h = relu(agg(x @ W1) + b1) ---
    gemm_wmma_f32_kernel<<<gGemm, blk, 0, stream>>>(x, W1, bufA, N, 256, 8);
    init_scaled_bias_kernel<<<gT, blk, 0, stream>>>(bufB, bufA, dinv, b1, total);
    edge_scatter_kernel<<<gScat, blk, 0, stream>>>(bufB, bufA, src, dst, dinv, E);
    relu_kernel<<<gT, blk, 0, stream>>>(bufB, total);

    // --- mu = agg(h @ W_mu) + b_mu ---
    gemm_wmma_f32_kernel<<<gGemm, blk, 0, stream>>>(bufB, Wmu, bufA, N, 128, 7);
    init_scaled_bias_kernel<<<gT, blk, 0, stream>>>(mu, bufA, dinv, bmu, total);
    edge_scatter_kernel<<<gScat, blk, 0, stream>>>(mu, bufA, src, dst, dinv, E);

    // --- logstd = agg(h @ W_ls) + b_ls ---
    gemm_wmma_f32_kernel<<<gGemm, blk, 0, stream>>>(bufB, Wls, bufA, N, 128, 7);
    init_scaled_bias_kernel<<<gT, blk, 0, stream>>>(ls, bufA, dinv, bls, total);
    edge_scatter_kernel<<<gScat, blk, 0, stream>>>(ls, bufA, src, dst, dinv, E);
}